// Model_1_0_34153579938563
// MI455X (gfx1250) — compile-verified
//
#include <hip/hip_runtime.h>

typedef __attribute__((ext_vector_type(2))) float v2f;
typedef __attribute__((ext_vector_type(8))) float v8f;

__device__ __forceinline__ float leaky(float v) { return v > 0.0f ? v : 0.01f * v; }

// ---------------------------------------------------------------------------
// deg[dst] += ew  (self-loop +1 added in dinv kernel)
// ---------------------------------------------------------------------------
__global__ void deg_accum_kernel(const int* __restrict__ dst, const float* __restrict__ ew,
                                 float* __restrict__ deg, int E) {
    int e = blockIdx.x * blockDim.x + threadIdx.x;
    if (e < E) atomicAdd(&deg[dst[e]], ew[e]);
}

__global__ void dinv_kernel(float* __restrict__ deg_dinv, int N) {
    int i = blockIdx.x * blockDim.x + threadIdx.x;
    if (i < N) deg_dinv[i] = rsqrtf(deg_dinv[i] + 1.0f);   // always > 0 due to self-loop
}

// ---------------------------------------------------------------------------
// H[N,16] = X[N,3] @ W[3,16]  -- one wave per 16-node tile, K padded 3->4.
// V_WMMA_F32_16X16X4_F32: A 16x4, B 4x16, C/D 16x16 f32.
// A layout: lanes 0-15 hold K=0/1 (V0/V1), lanes 16-31 hold K=2/3.
// D layout: VGPR r -> row m0 + r + 8*(lane>=16), col lane&15.
// ---------------------------------------------------------------------------
__global__ void gemm_in3_kernel(const float* __restrict__ X, const float* __restrict__ W,
                                float* __restrict__ H, int N) {
    int wave = (int)((blockIdx.x * blockDim.x + threadIdx.x) >> 5);
    int lane = threadIdx.x & 31;
    int m0 = wave << 4;
    if (m0 >= N) return;                 // wave-uniform: EXEC stays all-1s for WMMA
    int half = lane >> 4;                // 0: K={0,1}, 1: K={2,3}
    int lid  = lane & 15;
    int row  = m0 + lid;

    v2f A, B;
    if (half == 0) {
        A.x = X[(size_t)row * 3 + 0]; A.y = X[(size_t)row * 3 + 1];
        B.x = W[0 * 16 + lid];        B.y = W[1 * 16 + lid];
    } else {
        A.x = X[(size_t)row * 3 + 2]; A.y = 0.0f;   // K=3 zero pad
        B.x = W[2 * 16 + lid];        B.y = 0.0f;
    }
    v8f C = {};
    C = __builtin_amdgcn_wmma_f32_16x16x4_f32(false, A, false, B, (short)0, C, false, false);

    int mb = m0 + half * 8;
#pragma unroll
    for (int r = 0; r < 8; ++r) H[(size_t)(mb + r) * 16 + lid] = C[r];
}

// ---------------------------------------------------------------------------
// H[N,16] = IN[N,16] @ W[16,16] -- 4 chained K=4 WMMAs per wave tile.
// ---------------------------------------------------------------------------
__global__ void gemm_in16_kernel(const float* __restrict__ IN, const float* __restrict__ W,
                                 float* __restrict__ H, int N) {
    int wave = (int)((blockIdx.x * blockDim.x + threadIdx.x) >> 5);
    int lane = threadIdx.x & 31;
    int m0 = wave << 4;
    if (m0 >= N) return;
    int half = lane >> 4;
    int lid  = lane & 15;
    int row  = m0 + lid;

    v8f C = {};
#pragma unroll
    for (int kc = 0; kc < 4; ++kc) {
        int k = kc * 4 + half * 2;       // lanes 0-15: K=4kc,4kc+1; lanes 16-31: 4kc+2,4kc+3
        v2f A, B;
        A.x = IN[(size_t)row * 16 + k];
        A.y = IN[(size_t)row * 16 + k + 1];
        B.x = W[(size_t)k * 16 + lid];
        B.y = W[(size_t)(k + 1) * 16 + lid];
        C = __builtin_amdgcn_wmma_f32_16x16x4_f32(false, A, false, B, (short)0, C, false, false);
    }
    int mb = m0 + half * 8;
#pragma unroll
    for (int r = 0; r < 8; ++r) H[(size_t)(mb + r) * 16 + lid] = C[r];
}

// ---------------------------------------------------------------------------
// agg[dst] += norm_e * h[src]; 4 threads per edge, 4 features each.
// float4 gather -> single global_load_b128 per thread; atomics pipeline.
// ---------------------------------------------------------------------------
__global__ void edge_scatter_kernel(const int* __restrict__ src, const int* __restrict__ dst,
                                    const float* __restrict__ ew, const float* __restrict__ dinv,
                                    const float* __restrict__ H, float* __restrict__ agg, int E) {
    int tid = blockIdx.x * blockDim.x + threadIdx.x;
    int e = tid >> 2;
    if (e >= E) return;
    int q = tid & 3;
    int s = src[e], d = dst[e];
    float norm = dinv[s] * ew[e] * dinv[d];
    float4 hv = ((const float4*)(H + (size_t)s * 16))[q];    // global_load_b128
    float* ad = agg + (size_t)d * 16 + q * 4;
    atomicAdd(&ad[0], hv.x * norm);
    atomicAdd(&ad[1], hv.y * norm);
    atomicAdd(&ad[2], hv.z * norm);
    atomicAdd(&ad[3], hv.w * norm);
}

// out (in-place in agg) = leaky(agg + (1/deg)*h + b); one thread per (node, 4-feature quarter)
__global__ void finalize_kernel(const float* __restrict__ H, const float* __restrict__ dinv,
                                const float* __restrict__ b, float* __restrict__ agg, int N) {
    int tid = blockIdx.x * blockDim.x + threadIdx.x;
    if (tid >= N * 4) return;
    int i = tid >> 2, q = tid & 3;
    float di = dinv[i];
    float di2 = di * di;
    float4 av = ((const float4*)(agg + (size_t)i * 16))[q];
    float4 hv = ((const float4*)(H + (size_t)i * 16))[q];
    float4 bv = ((const float4*)b)[q];
    float4 o;
    o.x = leaky(av.x + di2 * hv.x + bv.x);
    o.y = leaky(av.y + di2 * hv.y + bv.y);
    o.z = leaky(av.z + di2 * hv.z + bv.z);
    o.w = leaky(av.w + di2 * hv.w + bv.w);
    ((float4*)(agg + (size_t)i * 16))[q] = o;
}

// Layer-2 finalize fused with mean-pool accumulation (one thread per node).
__global__ void finalize_pool_kernel(const float* __restrict__ H, const float* __restrict__ dinv,
                                     const float* __restrict__ b, const float* __restrict__ agg,
                                     const int* __restrict__ batch,
                                     float* __restrict__ sums, float* __restrict__ cnt, int N) {
    int i = blockIdx.x * blockDim.x + threadIdx.x;
    if (i >= N) return;
    float di = dinv[i];
    float di2 = di * di;
    int g = batch[i];
    float* sg = sums + (size_t)g * 16;
    const float4* av4 = (const float4*)(agg + (size_t)i * 16);
    const float4* hv4 = (const float4*)(H + (size_t)i * 16);
    const float4* bv4 = (const float4*)b;
#pragma unroll
    for (int q = 0; q < 4; ++q) {
        float4 av = av4[q], hv = hv4[q], bv = bv4[q];
        atomicAdd(&sg[q * 4 + 0], leaky(av.x + di2 * hv.x + bv.x));
        atomicAdd(&sg[q * 4 + 1], leaky(av.y + di2 * hv.y + bv.y));
        atomicAdd(&sg[q * 4 + 2], leaky(av.z + di2 * hv.z + bv.z));
        atomicAdd(&sg[q * 4 + 3], leaky(av.w + di2 * hv.w + bv.w));
    }
    atomicAdd(&cnt[g], 1.0f);
}

// ---------------------------------------------------------------------------
// Per-graph MLP heads (tiny: ~2 MFLOP total).
// ---------------------------------------------------------------------------
__device__ __forceinline__ void dense16(const float* in, const float* __restrict__ Wg,
                                        const float* __restrict__ bg, float* out) {
#pragma unroll
    for (int o = 0; o < 16; ++o) {
        float acc = bg[o];
#pragma unroll
        for (int k = 0; k < 16; ++k) acc += in[k] * Wg[k * 16 + o];
        out[o] = leaky(acc);
    }
}

__global__ void mlp_kernel(const float* __restrict__ sums, const float* __restrict__ cnt,
                           const float* __restrict__ Wp1, const float* __restrict__ bp1,
                           const float* __restrict__ Wp2, const float* __restrict__ bp2,
                           const float* __restrict__ Wp3, const float* __restrict__ bp3,
                           const float* __restrict__ Wt1, const float* __restrict__ bt1,
                           const float* __restrict__ Wt2, const float* __restrict__ bt2,
                           const float* __restrict__ Wt3, const float* __restrict__ bt3,
                           float* __restrict__ out, int G) {
    int g = blockIdx.x * blockDim.x + threadIdx.x;
    if (g >= G) return;
    float pooled[16], t1[16], t2[16];
    float c = fmaxf(cnt[g], 1.0f);
    float inv = 1.0f / c;
#pragma unroll
    for (int q = 0; q < 4; ++q) {
        float4 sv = ((const float4*)(sums + (size_t)g * 16))[q];
        pooled[q * 4 + 0] = sv.x * inv;
        pooled[q * 4 + 1] = sv.y * inv;
        pooled[q * 4 + 2] = sv.z * inv;
        pooled[q * 4 + 3] = sv.w * inv;
    }

    dense16(pooled, Wp1, bp1, t1);
    dense16(t1, Wp2, bp2, t2);
#pragma unroll
    for (int o = 0; o < 2; ++o) {
        float acc = bp3[o];
#pragma unroll
        for (int k = 0; k < 16; ++k) acc += t2[k] * Wp3[k * 2 + o];
        out[(size_t)g * 4 + o] = acc;
    }
    dense16(pooled, Wt1, bt1, t1);
    dense16(t1, Wt2, bt2, t2);
#pragma unroll
    for (int o = 0; o < 2; ++o) {
        float acc = bt3[o];
#pragma unroll
        for (int k = 0; k < 16; ++k) acc += t2[k] * Wt3[k * 2 + o];
        out[(size_t)g * 4 + 2 + o] = acc;
    }
}

// ---------------------------------------------------------------------------
extern "C" void kernel_launch(void* const* d_in, const int* in_sizes, int n_in,
                              void* d_out, int out_size, void* d_ws, size_t ws_size,
                              hipStream_t stream) {
    const float* X   = (const float*)d_in[0];
    const int*   ei  = (const int*)d_in[1];    // [2, E] int32
    const float* ew  = (const float*)d_in[2];
    const int*   bat = (const int*)d_in[3];
    const float* W1  = (const float*)d_in[5];
    const float* b1  = (const float*)d_in[6];
    const float* W2  = (const float*)d_in[7];
    const float* b2  = (const float*)d_in[8];
    const float* Wp1 = (const float*)d_in[9];
    const float* bp1 = (const float*)d_in[10];
    const float* Wp2 = (const float*)d_in[11];
    const float* bp2 = (const float*)d_in[12];
    const float* Wp3 = (const float*)d_in[13];
    const float* bp3 = (const float*)d_in[14];
    const float* Wt1 = (const float*)d_in[15];
    const float* bt1 = (const float*)d_in[16];
    const float* Wt2 = (const float*)d_in[17];
    const float* bt2 = (const float*)d_in[18];
    const float* Wt3 = (const float*)d_in[19];
    const float* bt3 = (const float*)d_in[20];
    float* out = (float*)d_out;

    const int N = in_sizes[0] / 3;
    const int E = in_sizes[2];
    const int G = out_size / 4;
    const int* src = ei;
    const int* dst = ei + E;

    // Workspace layout (floats): zeroed region first, h (no zeroing needed) last.
    //   dinv[N] | aggA[N*16] | aggB[N*16] | sums[G*16] | cnt[G] | h[N*16]
    float* ws   = (float*)d_ws;
    float* dinv = ws;
    float* aggA = dinv + N;
    float* aggB = aggA + (size_t)N * 16;
    float* sums = aggB + (size_t)N * 16;
    float* cnt  = sums + (size_t)G * 16;
    float* h    = cnt + G;
    size_t zero_bytes = ((size_t)N * 33 + (size_t)G * 17) * sizeof(float);
    hipMemsetAsync(d_ws, 0, zero_bytes, stream);

    const int B = 256;
    // 1) degree + normalization
    deg_accum_kernel<<<(E + B - 1) / B, B, 0, stream>>>(dst, ew, dinv, E);
    dinv_kernel<<<(N + B - 1) / B, B, 0, stream>>>(dinv, N);

    // 2) layer 1: h = X @ W1 (WMMA), scatter, finalize(+b1, leaky) -> aggA holds out1
    gemm_in3_kernel<<<(N + 127) / 128, B, 0, stream>>>(X, W1, h, N);
    edge_scatter_kernel<<<((E * 4) + B - 1) / B, B, 0, stream>>>(src, dst, ew, dinv, h, aggA, E);
    finalize_kernel<<<((N * 4) + B - 1) / B, B, 0, stream>>>(h, dinv, b1, aggA, N);

    // 3) layer 2: h = out1 @ W2 (WMMA), scatter into aggB, fused finalize + mean-pool accum
    gemm_in16_kernel<<<(N + 127) / 128, B, 0, stream>>>(aggA, W2, h, N);
    edge_scatter_kernel<<<((E * 4) + B - 1) / B, B, 0, stream>>>(src, dst, ew, dinv, h, aggB, E);
    finalize_pool_kernel<<<(N + B - 1) / B, B, 0, stream>>>(h, dinv, b2, aggB, bat, sums, cnt, N);

    // 4) per-graph MLP heads -> out[G,4]
    mlp_kernel<<<(G + B - 1) / B, B, 0, stream>>>(sums, cnt,
                                                  Wp1, bp1, Wp2, bp2, Wp3, bp3,
                                                  Wt1, bt1, Wt2, bt2, Wt3, bt3,
                                                  out, G);
}